// LabelSmoothKLDivLoss_87419764343430
// MI455X (gfx1250) — compile-verified
//
#include <hip/hip_runtime.h>
#include <hip/hip_bf16.h>
#include <stdint.h>

// Label-smoothing KLDiv loss, N=4096 rows x NCLASS=32000 (f32), reduction 'sum' / N.
//
// Per row:  C - [ PB*(S - NC*L) + (0.9-PB)*(x_y - L) ],  L = M + ln(sum exp(x-M))
//   where C = 0.1*ln(PB) + 0.9*ln(0.9)  (constant), M = row max, S = row sum,
//   PB = 0.1/31999.
//
// Roofline: 524 MB read once -> ~22.5 us at 23.3 TB/s. One v_exp_f32 per element
// (131M total) stays under the memory bound only with a two-pass softmax, so we
// stage the full 125 KB row in LDS (fits CDNA5's 320 KB/WGP) with
// global_load_async_to_lds_b128 (ASYNCcnt). The async ops apply IOFFSET to both
// the LDS and global address, so one base pair + offset: immediates covers the
// whole row; pass 1 is software-pipelined against the fill via in-order
// progressive s_wait_asynccnt.

#define NCLASS   32000
#define NROWS    4096
#define BLOCK_T  256
#define ROW_F4   (NCLASS / 4)                        // 8000 float4 chunks per row
#define ITERS    ((ROW_F4 + BLOCK_T - 1) / BLOCK_T)  // 32 (last iter partial)

static constexpr double PB_D   = 0.1 / 31999.0;
static constexpr double CTGT_D = -1.362428966520786;   // 0.1*ln(PB) + 0.9*ln(0.9)
static constexpr float  LOG2E  = 1.4426950408889634f;
static constexpr float  LN2    = 0.6931471805599453f;

__global__ void __launch_bounds__(BLOCK_T)
ls_kldiv_row_kernel(const float* __restrict__ x,
                    const int*   __restrict__ y,
                    float*       __restrict__ row_out) {
    extern __shared__ float smem[];          // [NCLASS] row  + [32] reduction scratch
    float* srow = smem;
    float* sred = smem + NCLASS;

    const int row = blockIdx.x;
    const int tid = threadIdx.x;
    const uint64_t rowbase =
        (uint64_t)(uintptr_t)(x + (size_t)row * (size_t)NCLASS);

    // ---- Stage the row into LDS with CDNA5 async global->LDS copies --------
    // IOFFSET is added to BOTH the LDS and global address (CDNA5 ISA
    // pseudocode), and our LDS image mirrors the global row, so a single base
    // address pair + offset:k*4096 immediates issues the whole row with no
    // per-iteration VALU address math.
    {
        // local->flat cast puts the LDS byte offset in the low 32 bits.
        uint32_t lds0 = (uint32_t)(uintptr_t)(&srow[tid * 4]);
        uint64_t ga0  = rowbase + (uint64_t)tid * 16u;
        #pragma unroll
        for (int k = 0; k < ITERS - 1; ++k) {          // k = 0..30, all lanes valid
            asm volatile("global_load_async_to_lds_b128 %0, %1, off offset:%2"
                         :: "v"(lds0), "v"(ga0), "i"(k * BLOCK_T * 16) : "memory");
        }
        // Tail chunk: only 64 lanes in range. Clamp so EVERY wave issues exactly
        // ITERS async ops (uniform ASYNCcnt for the progressive waits below);
        // out-of-range lanes rewrite the last chunk with identical data (benign).
        int idx4c = (ITERS - 1) * BLOCK_T + tid;
        if (idx4c > ROW_F4 - 1) idx4c = ROW_F4 - 1;
        uint32_t ldsT = (uint32_t)(uintptr_t)(&srow[idx4c * 4]);
        uint64_t gaT  = rowbase + (uint64_t)idx4c * 16u;
        asm volatile("global_load_async_to_lds_b128 %0, %1, off"
                     :: "v"(ldsT), "v"(gaT) : "memory");
    }

    const int wave = tid >> 5, lane = tid & 31;

    // ---- Pass 1: row max M and row sum S, pipelined against the LDS fill ---
    // Async loads complete in order per wave: after s_wait_asynccnt (31-k) the
    // first k+1 chunks are resident.
    float lm = -__builtin_inff();
    float ls = 0.0f;
    #pragma unroll
    for (int k = 0; k < ITERS; ++k) {
        asm volatile("s_wait_asynccnt %0" :: "i"(ITERS - 1 - k) : "memory");
        int idx4 = k * BLOCK_T + tid;
        if (idx4 < ROW_F4) {
            float4 v = *reinterpret_cast<const float4*>(&srow[idx4 * 4]);
            lm = fmaxf(lm, fmaxf(fmaxf(v.x, v.y), fmaxf(v.z, v.w)));
            ls += (v.x + v.y) + (v.z + v.w);
        }
    }
    #pragma unroll
    for (int off = 16; off > 0; off >>= 1) {
        lm = fmaxf(lm, __shfl_xor(lm, off, 32));
        ls += __shfl_xor(ls, off, 32);
    }
    if (lane == 0) { sred[wave] = lm; sred[8 + wave] = ls; }
    __syncthreads();
    if (tid == 0) {
        float M = sred[0], S = sred[8];
        #pragma unroll
        for (int w = 1; w < 8; ++w) { M = fmaxf(M, sred[w]); S += sred[8 + w]; }
        sred[16] = M;
        sred[17] = S;
    }
    __syncthreads();
    const float M = sred[16];

    // ---- Pass 2: sum exp(x - M), one v_exp_f32 per element -----------------
    float le = 0.0f;
    #pragma unroll
    for (int k = 0; k < ITERS; ++k) {
        int idx4 = k * BLOCK_T + tid;
        if (idx4 < ROW_F4) {
            float4 v = *reinterpret_cast<const float4*>(&srow[idx4 * 4]);
            le += __builtin_amdgcn_exp2f((v.x - M) * LOG2E)
                + __builtin_amdgcn_exp2f((v.y - M) * LOG2E)
                + __builtin_amdgcn_exp2f((v.z - M) * LOG2E)
                + __builtin_amdgcn_exp2f((v.w - M) * LOG2E);
        }
    }
    #pragma unroll
    for (int off = 16; off > 0; off >>= 1) le += __shfl_xor(le, off, 32);
    if (lane == 0) sred[wave] = le;      // reuse slots 0..7 (S stays at 17)
    __syncthreads();

    if (tid == 0) {
        float E = sred[0];
        #pragma unroll
        for (int w = 1; w < 8; ++w) E += sred[w];
        const float  S  = sred[17];
        const float  L  = M + __builtin_amdgcn_logf(E) * LN2;   // logsumexp
        const float  xy = srow[y[row]];
        // once-per-row combine in f64 (NC*L ~ 3.5e5; keep full precision)
        double cross = PB_D * ((double)S - (double)NCLASS * (double)L)
                     + (0.9 - PB_D) * ((double)xy - (double)L);
        row_out[row] = (float)((CTGT_D - cross) * (1.0 / (double)NROWS));
    }
}

// Deterministic final reduction of the 4096 per-row partials.
__global__ void __launch_bounds__(BLOCK_T)
ls_kldiv_final_kernel(const float* __restrict__ row_out, float* __restrict__ out) {
    __shared__ float s[8];
    const int tid = threadIdx.x;
    float a = 0.0f;
    #pragma unroll
    for (int k = 0; k < NROWS / BLOCK_T; ++k) a += row_out[k * BLOCK_T + tid];
    #pragma unroll
    for (int off = 16; off > 0; off >>= 1) a += __shfl_xor(a, off, 32);
    if ((tid & 31) == 0) s[tid >> 5] = a;
    __syncthreads();
    if (tid == 0) {
        float t = 0.0f;
        #pragma unroll
        for (int w = 0; w < 8; ++w) t += s[w];
        out[0] = t;
    }
}

extern "C" void kernel_launch(void* const* d_in, const int* in_sizes, int n_in,
                              void* d_out, int out_size, void* d_ws, size_t ws_size,
                              hipStream_t stream) {
    const float* x   = (const float*)d_in[0];
    const int*   y   = (const int*)d_in[1];
    float*       ws  = (float*)d_ws;             // 4096 per-row partials
    float*       out = (float*)d_out;

    const size_t shmem = (size_t)(NCLASS + 32) * sizeof(float);  // 128,128 B dynamic LDS
    ls_kldiv_row_kernel<<<NROWS, BLOCK_T, shmem, stream>>>(x, y, ws);
    ls_kldiv_final_kernel<<<1, BLOCK_T, 0, stream>>>(ws, out);
}